// GraphNodeEncoder_84817014162125
// MI455X (gfx1250) — compile-verified
//
#include <hip/hip_runtime.h>
#include <cstdint>
#include <cstddef>

typedef float v2f __attribute__((ext_vector_type(2)));
typedef float v8f __attribute__((ext_vector_type(8)));

// ---------------------------------------------------------------------------
// f32 WMMA GEMM: C[M,N] = A[M,K] @ B[K,N] (+bias) (optional ReLU)
// block = 128 threads (4 waves), tile = 64x64, K chunk = 32.
// Uses V_WMMA_F32_16X16X4_F32 (CDNA5).
// ---------------------------------------------------------------------------
#define BM 64
#define BN 64
#define KT 32

__global__ __launch_bounds__(128) void gemm_wmma_kernel(
    const float* __restrict__ A, const float* __restrict__ B,
    const float* __restrict__ bias, float* __restrict__ C,
    int M, int N, int K, int do_relu)
{
  __shared__ float As[BM][KT];       // 8 KB
  __shared__ float Bs[BN][KT + 2];   // 8.5 KB, even-padded so v2f LDS reads are 8B aligned

  const int tid  = threadIdx.x;
  const int wave = tid >> 5;
  const int lane = tid & 31;
  const int tile_m = blockIdx.y * BM;
  const int tile_n = blockIdx.x * BN;

  const v8f vzero = {0.f, 0.f, 0.f, 0.f, 0.f, 0.f, 0.f, 0.f};
  v8f acc[4];
  #pragma unroll
  for (int j = 0; j < 4; ++j) acc[j] = vzero;

  // A-fragment mapping (16x4 f32): lanes 0-15 -> K pair {0,1}; lanes 16-31 -> {2,3}
  const int mrow = wave * 16 + (lane & 15);
  const int koff = (lane >> 4) * 2;

  for (int k0 = 0; k0 < K; k0 += KT) {
    __syncthreads();
    // Stage A tile (BM x KT), zero rows beyond M
    for (int i = tid; i < BM * KT / 4; i += 128) {
      int r  = i >> 3;              // KT/4 == 8 float4 per row
      int c4 = (i & 7) << 2;
      int gr = tile_m + r;
      float4 v = make_float4(0.f, 0.f, 0.f, 0.f);
      if (gr < M) v = *(const float4*)(A + (size_t)gr * K + k0 + c4);
      *(float4*)&As[r][c4] = v;
    }
    // Stage B tile (KT x BN) transposed into Bs[n][k]
    for (int i = tid; i < KT * BN / 4; i += 128) {
      int r  = i >> 4;              // k row 0..31
      int c4 = (i & 15) << 2;       // n offset
      float4 v = *(const float4*)(B + (size_t)(k0 + r) * N + tile_n + c4);
      Bs[c4 + 0][r] = v.x;
      Bs[c4 + 1][r] = v.y;
      Bs[c4 + 2][r] = v.z;
      Bs[c4 + 3][r] = v.w;
    }
    __syncthreads();

    if (k0 + KT < K) {
      // warm caches for next K-chunk of A (lowers to global_prefetch_b8)
      int gr = tile_m + (tid >> 1);
      if (gr < M)
        __builtin_prefetch(A + (size_t)gr * K + k0 + KT + (tid & 1) * (KT / 2), 0, 0);
    }

    #pragma unroll
    for (int kk = 0; kk < KT; kk += 4) {
      v2f a = *(const v2f*)&As[mrow][kk + koff];
      #pragma unroll
      for (int j = 0; j < 4; ++j) {
        int ncol = j * 16 + (lane & 15);
        v2f b = *(const v2f*)&Bs[ncol][kk + koff];
        acc[j] = __builtin_amdgcn_wmma_f32_16x16x4_f32(
            false, a, false, b, (short)0, acc[j], false, false);
      }
    }
  }

  // C/D layout: VGPR v -> M = v (+8 for lanes 16-31), N = lane%16
  const int mbase = tile_m + wave * 16 + (lane >> 4) * 8;
  const int nb = lane & 15;
  #pragma unroll
  for (int j = 0; j < 4; ++j) {
    int n = tile_n + j * 16 + nb;
    float bv = bias ? bias[n] : 0.f;
    #pragma unroll
    for (int v = 0; v < 8; ++v) {
      int m = mbase + v;
      if (m < M) {
        float val = acc[j][v] + bv;
        if (do_relu) val = fmaxf(val, 0.f);
        C[(size_t)m * N + n] = val;
      }
    }
  }
}

// ---------------------------------------------------------------------------
// Elementwise / edge kernels
// ---------------------------------------------------------------------------
__global__ void fill_kernel(float* __restrict__ p, float v, size_t n)
{
  size_t i = blockIdx.x * (size_t)blockDim.x + threadIdx.x;
  size_t stride = (size_t)gridDim.x * blockDim.x;
  for (; i < n; i += stride) p[i] = v;
}

// al[n*H+h] = sum_c h[(n*H+h)*C + c] * att[h*C + c]
__global__ void att_logits_kernel(const float* __restrict__ h,
                                  const float* __restrict__ att,
                                  float* __restrict__ al,
                                  int Nn, int Hh, int Cc)
{
  int t = blockIdx.x * blockDim.x + threadIdx.x;
  if (t >= Nn * Hh) return;
  int hh = t % Hh;
  const float* hp = h + (size_t)t * Cc;
  const float* ap = att + (size_t)hh * Cc;
  float s = 0.f;
  for (int c = 0; c < Cc; ++c) s += hp[c] * ap[c];
  al[t] = s;
}

__device__ __forceinline__ float leaky02(float v) { return v > 0.f ? v : 0.2f * v; }

__device__ __forceinline__ void atomic_max_f32(float* addr, float val)
{
  int* ia = (int*)addr;
  int old = *ia;
  while (__int_as_float(old) < val) {
    int assumed = old;
    old = atomicCAS(ia, assumed, __float_as_int(val));
    if (old == assumed) break;
  }
}

__global__ void edge_max_kernel(const int* __restrict__ src, const int* __restrict__ dst,
                                const float* __restrict__ als, const float* __restrict__ ald,
                                float* __restrict__ m, int E, int Hh)
{
  int t = blockIdx.x * blockDim.x + threadIdx.x;
  if (t >= E * Hh) return;
  int e = t / Hh, hh = t - e * Hh;
  int s = src[e], d = dst[e];
  float v = leaky02(als[(size_t)s * Hh + hh] + ald[(size_t)d * Hh + hh]);
  atomic_max_f32(&m[(size_t)d * Hh + hh], v);
}

__global__ void edge_expsum_kernel(const int* __restrict__ src, const int* __restrict__ dst,
                                   const float* __restrict__ als, const float* __restrict__ ald,
                                   const float* __restrict__ m,
                                   float* __restrict__ ebuf, float* __restrict__ ssum,
                                   int E, int Hh)
{
  int t = blockIdx.x * blockDim.x + threadIdx.x;
  if (t >= E * Hh) return;
  int e = t / Hh, hh = t - e * Hh;
  int s = src[e], d = dst[e];
  float v = leaky02(als[(size_t)s * Hh + hh] + ald[(size_t)d * Hh + hh]);
  float ex = expf(v - m[(size_t)d * Hh + hh]);
  ebuf[t] = ex;
  atomicAdd(&ssum[(size_t)d * Hh + hh], ex);
}

// one thread per (edge, head, channel): coalesced gather + scatter-add
__global__ void edge_scatter_kernel(const int* __restrict__ src, const int* __restrict__ dst,
                                    const float* __restrict__ ebuf, const float* __restrict__ ssum,
                                    const float* __restrict__ hsrc, float* __restrict__ agg,
                                    int E, int Hh, int Cc)
{
  size_t t = blockIdx.x * (size_t)blockDim.x + threadIdx.x;
  size_t total = (size_t)E * Hh * Cc;
  if (t >= total) return;
  int c = (int)(t % Cc);
  size_t eh = t / Cc;
  int hh = (int)(eh % Hh);
  int e  = (int)(eh / Hh);
  int s = src[e], d = dst[e];
  float alpha = ebuf[eh] / (ssum[(size_t)d * Hh + hh] + 1e-16f);
  atomicAdd(&agg[((size_t)d * Hh + hh) * Cc + c],
            alpha * hsrc[((size_t)s * Hh + hh) * Cc + c]);
}

__global__ void bias_relu_kernel(float* __restrict__ x, const float* __restrict__ bias,
                                 size_t total, int W)
{
  size_t t = blockIdx.x * (size_t)blockDim.x + threadIdx.x;
  if (t >= total) return;
  int c = (int)(t % W);
  x[t] = fmaxf(x[t] + bias[c], 0.f);
}

// ---------------------------------------------------------------------------
// Launcher
// ---------------------------------------------------------------------------
extern "C" void kernel_launch(void* const* d_in, const int* in_sizes, int n_in,
                              void* d_out, int out_size, void* d_ws, size_t ws_size,
                              hipStream_t stream)
{
  const float* x_t   = (const float*)d_in[0];
  const float* x_p   = (const float*)d_in[1];
  const int*   e_tp  = (const int*)d_in[2];
  const int*   e_pt  = (const int*)d_in[3];
  const float* W1_tp = (const float*)d_in[4];
  const float* as1_tp= (const float*)d_in[5];
  const float* ad1_tp= (const float*)d_in[6];
  const float* b1_tp = (const float*)d_in[7];
  const float* W1_pt = (const float*)d_in[8];
  const float* as1_pt= (const float*)d_in[9];
  const float* ad1_pt= (const float*)d_in[10];
  const float* b1_pt = (const float*)d_in[11];
  const float* W2_tp = (const float*)d_in[12];
  const float* as2_tp= (const float*)d_in[13];
  const float* ad2_tp= (const float*)d_in[14];
  const float* b2_tp = (const float*)d_in[15];
  const float* W2_pt = (const float*)d_in[16];
  const float* as2_pt= (const float*)d_in[17];
  const float* ad2_pt= (const float*)d_in[18];
  const float* b2_pt = (const float*)d_in[19];
  const float* Wo_t  = (const float*)d_in[20];
  const float* bo_t  = (const float*)d_in[21];
  const float* Wo_p  = (const float*)d_in[22];
  const float* bo_p  = (const float*)d_in[23];

  const int HC  = in_sizes[7];            // H*HID = 512
  const int HID = in_sizes[15];           // 64
  const int H   = HC / HID;               // 8
  const int D   = in_sizes[4] / HC;       // 128
  const int Nt  = in_sizes[0] / D;        // 50000
  const int Np  = in_sizes[1] / D;        // 50000
  const int Etp = in_sizes[2] / 2;        // 400000
  const int Ept = in_sizes[3] / 2;
  const int OUT = in_sizes[21];           // 128

  const int* src_tp = e_tp;            const int* dst_tp = e_tp + Etp;
  const int* src_pt = e_pt;            const int* dst_pt = e_pt + Ept;

  // ---- workspace layout (floats) ----
  float* ws = (float*)d_ws;
  size_t off = 0;
  auto alloc = [&](size_t nE) { float* p = ws + off; off += nE; return p; };
  float* h_src_tp = alloc((size_t)Nt * HC);   // x_t @ W1_tp  (persists through scatter)
  float* h_src_pt = alloc((size_t)Np * HC);   // x_p @ W1_pt
  float* tmp      = alloc((size_t)((Nt > Np ? Nt : Np)) * HC); // h_dst scratch (reused L2)
  float* agg_p    = alloc((size_t)Np * HC);   // -> p1
  float* agg_t    = alloc((size_t)Nt * HC);   // -> t1
  float* al_s_tp  = alloc((size_t)Nt * H);
  float* al_d_tp  = alloc((size_t)Np * H);
  float* al_s_pt  = alloc((size_t)Np * H);
  float* al_d_pt  = alloc((size_t)Nt * H);
  float* m_tp = alloc((size_t)Np * H);  float* s_tp = alloc((size_t)Np * H);
  float* m_pt = alloc((size_t)Nt * H);  float* s_pt = alloc((size_t)Nt * H);
  float* ebuf_tp = alloc((size_t)Etp * H);
  float* ebuf_pt = alloc((size_t)Ept * H);
  float* agg2_p   = alloc((size_t)Np * HID);  // -> p2
  float* agg2_t   = alloc((size_t)Nt * HID);  // -> t2
  float* h2_src_tp = alloc((size_t)Nt * HID); // t1 @ W2_tp
  float* h2_src_pt = alloc((size_t)Np * HID); // p1 @ W2_pt
  (void)ws_size; (void)n_in;

  auto gemm = [&](const float* A, const float* B, const float* bias, float* C,
                  int M, int N, int K, int relu) {
    dim3 grid(N / BN, (M + BM - 1) / BM);
    gemm_wmma_kernel<<<grid, 128, 0, stream>>>(A, B, bias, C, M, N, K, relu);
  };
  auto fill = [&](float* p, float v, size_t n) {
    size_t g = (n + 255) / 256; if (g > (1u << 20)) g = (1u << 20);
    fill_kernel<<<(int)g, 256, 0, stream>>>(p, v, n);
  };
  auto logits = [&](const float* h, const float* att, float* al, int Nn, int Hh, int Cc) {
    int total = Nn * Hh;
    att_logits_kernel<<<(total + 255) / 256, 256, 0, stream>>>(h, att, al, Nn, Hh, Cc);
  };
  auto gat_edges = [&](const int* src, const int* dst, int E,
                       const float* al_s, const float* al_d,
                       float* m, float* s, float* ebuf,
                       const float* hsrc, float* agg, int Nd, int Hh, int Cc) {
    fill(m, -3.0e38f, (size_t)Nd * Hh);
    fill(s, 0.f, (size_t)Nd * Hh);
    fill(agg, 0.f, (size_t)Nd * Hh * Cc);
    int tEH = E * Hh;
    edge_max_kernel<<<(tEH + 255) / 256, 256, 0, stream>>>(src, dst, al_s, al_d, m, E, Hh);
    edge_expsum_kernel<<<(tEH + 255) / 256, 256, 0, stream>>>(src, dst, al_s, al_d, m, ebuf, s, E, Hh);
    size_t tEHC = (size_t)tEH * Cc;
    edge_scatter_kernel<<<(int)((tEHC + 255) / 256), 256, 0, stream>>>(src, dst, ebuf, s, hsrc, agg, E, Hh, Cc);
  };
  auto bias_relu = [&](float* x, const float* b, int Nn, int W) {
    size_t total = (size_t)Nn * W;
    bias_relu_kernel<<<(int)((total + 255) / 256), 256, 0, stream>>>(x, b, total, W);
  };

  // ================= Layer 1 =================
  // dst projections (transient): only needed for al_dst
  gemm(x_p, W1_tp, nullptr, tmp, Np, HC, D, 0);
  logits(tmp, ad1_tp, al_d_tp, Np, H, HID);
  gemm(x_t, W1_pt, nullptr, tmp, Nt, HC, D, 0);
  logits(tmp, ad1_pt, al_d_pt, Nt, H, HID);
  // src projections (persist through scatter)
  gemm(x_t, W1_tp, nullptr, h_src_tp, Nt, HC, D, 0);
  logits(h_src_tp, as1_tp, al_s_tp, Nt, H, HID);
  gemm(x_p, W1_pt, nullptr, h_src_pt, Np, HC, D, 0);
  logits(h_src_pt, as1_pt, al_s_pt, Np, H, HID);
  // edge softmax + aggregate
  gat_edges(src_tp, dst_tp, Etp, al_s_tp, al_d_tp, m_tp, s_tp, ebuf_tp, h_src_tp, agg_p, Np, H, HID);
  gat_edges(src_pt, dst_pt, Ept, al_s_pt, al_d_pt, m_pt, s_pt, ebuf_pt, h_src_pt, agg_t, Nt, H, HID);
  bias_relu(agg_p, b1_tp, Np, HC);   // p1
  bias_relu(agg_t, b1_pt, Nt, HC);   // t1

  // ================= Layer 2 (heads=1, concat=False -> identity mean) =======
  // tp direction: src = t1, dst = p1
  gemm(agg_t, W2_tp, nullptr, h2_src_tp, Nt, HID, HC, 0);
  logits(h2_src_tp, as2_tp, al_s_tp, Nt, 1, HID);
  gemm(agg_p, W2_tp, nullptr, tmp, Np, HID, HC, 0);
  logits(tmp, ad2_tp, al_d_tp, Np, 1, HID);
  gat_edges(src_tp, dst_tp, Etp, al_s_tp, al_d_tp, m_tp, s_tp, ebuf_tp, h2_src_tp, agg2_p, Np, 1, HID);
  bias_relu(agg2_p, b2_tp, Np, HID); // p2
  // pt direction: src = p1, dst = t1
  gemm(agg_p, W2_pt, nullptr, h2_src_pt, Np, HID, HC, 0);
  logits(h2_src_pt, as2_pt, al_s_pt, Np, 1, HID);
  gemm(agg_t, W2_pt, nullptr, tmp, Nt, HID, HC, 0);
  logits(tmp, ad2_pt, al_d_pt, Nt, 1, HID);
  gat_edges(src_pt, dst_pt, Ept, al_s_pt, al_d_pt, m_pt, s_pt, ebuf_pt, h2_src_pt, agg2_t, Nt, 1, HID);
  bias_relu(agg2_t, b2_pt, Nt, HID); // t2

  // ================= Output linears =================
  float* out_t = (float*)d_out;
  float* out_p = out_t + (size_t)Nt * OUT;
  gemm(agg2_t, Wo_t, bo_t, out_t, Nt, OUT, HID, 0);
  gemm(agg2_p, Wo_p, bo_p, out_p, Np, OUT, HID, 0);
  (void)out_size;
}